// ModelName_13400297963844
// MI455X (gfx1250) — compile-verified
//
#include <hip/hip_runtime.h>
#include <hip/hip_bf16.h>

typedef __attribute__((ext_vector_type(2))) float v2f;
typedef __attribute__((ext_vector_type(8))) float v8f;

#define HID 64
#define NEG_SLOPE 0.2f

// =======================================================================
// GEMM via fp32 WMMA: H[M x 64] = X[M x K] @ W[K x 64]
// block = 128 threads (4 waves); wave w computes output cols [16w, 16w+16)
// grid.x = ceil(M/16). K must be a multiple of 4 (128 and 64 here).
//
// V_WMMA_F32_16X16X4_F32 operand layout (ISA 7.12.2):
//   A (16x4):  lanes 0-15 -> row M=lane, VGPR0/1 = K0,K1 ; lanes 16-31 -> K2,K3
//   B (4x16):  lanes 0-15 -> col N=lane, VGPR0/1 = K0,K1 ; lanes 16-31 -> K2,K3
//   C/D (16x16): VGPR r -> row = r + 8*(lane>=16), col = lane%16
// =======================================================================
__global__ __launch_bounds__(128)
void k_gemm_wmma(const float* __restrict__ X, const float* __restrict__ W,
                 float* __restrict__ H, int M, int K) {
  const int wave = threadIdx.x >> 5;
  const int lane = threadIdx.x & 31;
  const int half = lane >> 4;          // 0: K0/K1, 1: K2/K3
  const int l16  = lane & 15;
  const int mrow = blockIdx.x * 16 + l16;
  const int arow = mrow < M ? mrow : (M - 1);   // clamp so OOB tiles read safely
  const float* Xr = X + (long long)arow * K;
  const int col = (wave << 4) + l16;

  v8f c = {};
  for (int k0 = 0; k0 < K; k0 += 4) {
    const int ka = k0 + (half << 1);
    v2f a, b;
    a.x = Xr[ka];
    a.y = Xr[ka + 1];
    b.x = W[(long long)ka * HID + col];
    b.y = W[(long long)(ka + 1) * HID + col];
    // 8 args: (neg_a, A, neg_b, B, c_mod, C, reuse_a, reuse_b)
    c = __builtin_amdgcn_wmma_f32_16x16x4_f32(false, a, false, b,
                                              (short)0, c, false, false);
  }

  const int rbase = blockIdx.x * 16 + half * 8;
  float* Ho = H + (long long)rbase * HID + col;
#pragma unroll
  for (int r = 0; r < 8; ++r) {
    if (rbase + r < M) Ho[(long long)r * HID] = c[r];
  }
}

// =======================================================================
// Per-node attention scores: sc_s[i] = h[i].a_src ; sc_d[i] = h[i].a_dst
// One wave per node, 2 features per lane, shuffle reduction.
// =======================================================================
__global__ __launch_bounds__(256)
void k_scores(const float* __restrict__ h, const float* __restrict__ a_s,
              const float* __restrict__ a_d, float* __restrict__ sc_s,
              float* __restrict__ sc_d, int Nn) {
  const int gid  = blockIdx.x * blockDim.x + threadIdx.x;
  const int node = gid >> 5;
  const int lane = gid & 31;
  if (node >= Nn) return;
  float2 hv = ((const float2*)(h + (long long)node * HID))[lane];
  float2 sv = ((const float2*)a_s)[lane];
  float2 dv = ((const float2*)a_d)[lane];
  float ps = hv.x * sv.x + hv.y * sv.y;
  float pd = hv.x * dv.x + hv.y * dv.y;
#pragma unroll
  for (int o = 16; o > 0; o >>= 1) {
    ps += __shfl_down(ps, o, 32);
    pd += __shfl_down(pd, o, 32);
  }
  if (lane == 0) { sc_s[node] = ps; sc_d[node] = pd; }
}

// =======================================================================
// CSR-by-dst construction (built once per launch, reused by both layers)
// =======================================================================
__global__ __launch_bounds__(256)
void k_zero_i32(int* __restrict__ a, int n) {
  const int t = blockIdx.x * blockDim.x + threadIdx.x;
  if (t < n) a[t] = 0;
}

__device__ __forceinline__ int edge_dst(const int* __restrict__ ei, int E, int e) {
  return (e < E) ? ei[E + e] : (e - E);   // appended self loops: dst = e-E
}
__device__ __forceinline__ int edge_src(const int* __restrict__ ei, int E, int e) {
  return (e < E) ? ei[e] : (e - E);
}

__global__ __launch_bounds__(256)
void k_deg(const int* __restrict__ ei, int E, int Nn, int* __restrict__ deg) {
  const int t = blockIdx.x * blockDim.x + threadIdx.x;
  if (t >= E + Nn) return;
  atomicAdd(deg + edge_dst(ei, E, t), 1);
}

// single-block exclusive scan (Hillis-Steele per 1024-tile, serial carry)
__global__ __launch_bounds__(1024)
void k_scan(const int* __restrict__ deg, int* __restrict__ start,
            int* __restrict__ cur, int Nn) {
  __shared__ int sh[1024];
  __shared__ int carry;
  if (threadIdx.x == 0) carry = 0;
  __syncthreads();
  for (int base = 0; base < Nn; base += 1024) {
    const int i = base + (int)threadIdx.x;
    const int v = (i < Nn) ? deg[i] : 0;
    sh[threadIdx.x] = v;
    __syncthreads();
    for (int o = 1; o < 1024; o <<= 1) {
      const int t = (threadIdx.x >= (unsigned)o) ? sh[threadIdx.x - o] : 0;
      __syncthreads();
      sh[threadIdx.x] += t;
      __syncthreads();
    }
    const int excl = carry + sh[threadIdx.x] - v;
    if (i < Nn) { start[i] = excl; cur[i] = excl; }
    __syncthreads();
    if (threadIdx.x == 0) carry += sh[1023];
    __syncthreads();
  }
}

__global__ __launch_bounds__(256)
void k_scatter(const int* __restrict__ ei, int E, int Nn,
               int* __restrict__ cur, int* __restrict__ csr) {
  const int t = blockIdx.x * blockDim.x + threadIdx.x;
  if (t >= E + Nn) return;
  const int d = edge_dst(ei, E, t);
  const int slot = atomicAdd(cur + d, 1);
  csr[slot] = t;
}

// =======================================================================
// Per-dst segment softmax, atomic-free: one wave per dst node.
// Writes normalized alpha (CSR order) into ev.
// =======================================================================
__global__ __launch_bounds__(256)
void k_attn(const int* __restrict__ csr, const int* __restrict__ start,
            const int* __restrict__ deg, const int* __restrict__ ei, int E,
            const float* __restrict__ sc_s, const float* __restrict__ sc_d,
            float* __restrict__ ev, int Nn) {
  const int gid  = blockIdx.x * blockDim.x + threadIdx.x;
  const int d    = gid >> 5;
  const int lane = gid & 31;
  if (d >= Nn) return;
  const int s0 = start[d];
  const int n  = deg[d];
  const float sd = sc_d[d];

  float m = -__builtin_inff();
  for (int j = lane; j < n; j += 32) {
    const int e = csr[s0 + j];
    const int s = edge_src(ei, E, e);
    float v = sc_s[s] + sd;
    v = v < 0.0f ? v * NEG_SLOPE : v;
    ev[s0 + j] = v;
    m = fmaxf(m, v);
  }
#pragma unroll
  for (int o = 16; o > 0; o >>= 1) m = fmaxf(m, __shfl_xor(m, o, 32));

  float sum = 0.0f;
  for (int j = lane; j < n; j += 32) {
    const float ex = __expf(ev[s0 + j] - m);
    ev[s0 + j] = ex;
    sum += ex;
  }
#pragma unroll
  for (int o = 16; o > 0; o >>= 1) sum += __shfl_xor(sum, o, 32);

  const float inv = 1.0f / sum;   // every node has a self loop -> sum > 0
  for (int j = lane; j < n; j += 32) ev[s0 + j] *= inv;
}

// =======================================================================
// Per-dst weighted aggregation + bias (+ optional ReLU), atomic-free.
// One wave per dst; lanes own 2 features (float2); serial edge loop with
// coalesced 256B gathers of h[src].
// =======================================================================
__global__ __launch_bounds__(256)
void k_aggregate(const int* __restrict__ csr, const int* __restrict__ start,
                 const int* __restrict__ deg, const int* __restrict__ ei, int E,
                 const float* __restrict__ ev, const float* __restrict__ h,
                 const float* __restrict__ bias, float* __restrict__ out,
                 int Nn, int do_relu) {
  const int gid  = blockIdx.x * blockDim.x + threadIdx.x;
  const int d    = gid >> 5;
  const int lane = gid & 31;
  if (d >= Nn) return;
  const int s0 = start[d];
  const int n  = deg[d];

  float2 acc = make_float2(0.0f, 0.0f);
  for (int j = 0; j < n; ++j) {
    const int e = csr[s0 + j];          // wave-uniform broadcast loads
    const int s = edge_src(ei, E, e);
    const float a = ev[s0 + j];
    const float2 hv = ((const float2*)(h + (long long)s * HID))[lane];
    acc.x += a * hv.x;
    acc.y += a * hv.y;
  }
  const float2 bv = ((const float2*)bias)[lane];
  acc.x += bv.x;
  acc.y += bv.y;
  if (do_relu) { acc.x = fmaxf(acc.x, 0.0f); acc.y = fmaxf(acc.y, 0.0f); }
  ((float2*)(out + (long long)d * HID))[lane] = acc;
}

// =======================================================================
// Pool + head
// =======================================================================
__global__ __launch_bounds__(256)
void k_init_pool(float* __restrict__ gsum, float* __restrict__ gcnt, int G) {
  const int t = blockIdx.x * blockDim.x + threadIdx.x;
  if (t < G * HID) gsum[t] = 0.0f;
  if (t < G) gcnt[t] = 0.0f;
}

__global__ __launch_bounds__(256)
void k_pool(const float* __restrict__ h, const int* __restrict__ batch,
            float* __restrict__ gsum, float* __restrict__ gcnt, int Nn) {
  const int t = blockIdx.x * blockDim.x + threadIdx.x;
  if (t >= Nn * HID) return;
  const int n = t >> 6;
  const int f = t & 63;
  const int g = batch[n];
  atomicAdd(gsum + (long long)g * HID + f, h[t]);
  if (f == 0) atomicAdd(gcnt + g, 1.0f);
}

// per-graph head: g=sum/cnt; z=relu(g@lw+lb) [32]; logits=z@cw+cb [6]; log_softmax
__global__ __launch_bounds__(32)
void k_head(const float* __restrict__ gsum, const float* __restrict__ gcnt,
            const float* __restrict__ lw, const float* __restrict__ lb,
            const float* __restrict__ cw, const float* __restrict__ cb,
            float* __restrict__ out) {
  const int g = blockIdx.x;
  const int lane = threadIdx.x;
  __shared__ float gsh[HID];
  const float cnt = fmaxf(gcnt[g], 1.0f);
  gsh[lane]      = gsum[(long long)g * HID + lane] / cnt;
  gsh[lane + 32] = gsum[(long long)g * HID + lane + 32] / cnt;
  __syncthreads();
  float z = lb[lane];
  for (int k = 0; k < HID; ++k) z += gsh[k] * lw[k * 32 + lane];
  z = fmaxf(z, 0.0f);
  float logits[6];
#pragma unroll
  for (int c = 0; c < 6; ++c) {
    float p = z * cw[lane * 6 + c];
#pragma unroll
    for (int o = 16; o > 0; o >>= 1) p += __shfl_xor(p, o, 32);
    logits[c] = p + cb[c];
  }
  float m = logits[0];
#pragma unroll
  for (int c = 1; c < 6; ++c) m = fmaxf(m, logits[c]);
  float s = 0.0f;
#pragma unroll
  for (int c = 0; c < 6; ++c) s += __expf(logits[c] - m);
  const float lse = m + __logf(s);
  if (lane < 6) out[g * 6 + lane] = logits[lane] - lse;
}

// =======================================================================
extern "C" void kernel_launch(void* const* d_in, const int* in_sizes, int n_in,
                              void* d_out, int out_size, void* d_ws, size_t ws_size,
                              hipStream_t stream) {
  const float* x   = (const float*)d_in[0];
  const int*   ei  = (const int*)d_in[1];
  const int*  batch= (const int*)d_in[2];
  const float* W1  = (const float*)d_in[3];
  const float* as1 = (const float*)d_in[4];
  const float* ad1 = (const float*)d_in[5];
  const float* b1  = (const float*)d_in[6];
  const float* W2  = (const float*)d_in[7];
  const float* as2 = (const float*)d_in[8];
  const float* ad2 = (const float*)d_in[9];
  const float* b2  = (const float*)d_in[10];
  const float* lw  = (const float*)d_in[11];
  const float* lb  = (const float*)d_in[12];
  const float* cw  = (const float*)d_in[13];
  const float* cb  = (const float*)d_in[14];
  float* out = (float*)d_out;

  const int Nn = in_sizes[2];          // 100000 (batch vector length)
  const int E  = in_sizes[1] / 2;      // 3.2M
  const int K1 = in_sizes[0] / Nn;     // 128
  const int G  = out_size / 6;         // 512
  const int Etot = E + Nn;

  // bump-allocate scratch from d_ws (256B aligned chunks, ~80 MB total)
  char* p = (char*)d_ws;
  auto alloc = [&](size_t bytes) {
    char* r = p;
    p += (bytes + 255) & ~(size_t)255;
    return r;
  };
  float* hA   = (float*)alloc((size_t)Nn * HID * sizeof(float));
  float* hB   = (float*)alloc((size_t)Nn * HID * sizeof(float));
  float* scS  = (float*)alloc((size_t)Nn * sizeof(float));
  float* scD  = (float*)alloc((size_t)Nn * sizeof(float));
  float* ev   = (float*)alloc((size_t)Etot * sizeof(float));
  int*   deg  = (int*)  alloc((size_t)Nn * sizeof(int));
  int*   st   = (int*)  alloc((size_t)Nn * sizeof(int));
  int*   cur  = (int*)  alloc((size_t)Nn * sizeof(int));
  int*   csr  = (int*)  alloc((size_t)Etot * sizeof(int));
  float* gsum = (float*)alloc((size_t)G * HID * sizeof(float));
  float* gcnt = (float*)alloc((size_t)G * sizeof(float));

  auto cdiv = [](long long a, long long b) { return (unsigned)((a + b - 1) / b); };

  // ---------------- CSR build (once) ----------------
  k_zero_i32<<<cdiv(Nn, 256), 256, 0, stream>>>(deg, Nn);
  k_deg<<<cdiv(Etot, 256), 256, 0, stream>>>(ei, E, Nn, deg);
  k_scan<<<1, 1024, 0, stream>>>(deg, st, cur, Nn);
  k_scatter<<<cdiv(Etot, 256), 256, 0, stream>>>(ei, E, Nn, cur, csr);

  // ---------------- layer 1 ----------------
  k_gemm_wmma<<<cdiv(Nn, 16), 128, 0, stream>>>(x, W1, hA, Nn, K1);
  k_scores<<<cdiv((long long)Nn * 32, 256), 256, 0, stream>>>(hA, as1, ad1, scS, scD, Nn);
  k_attn<<<cdiv((long long)Nn * 32, 256), 256, 0, stream>>>(csr, st, deg, ei, E, scS, scD, ev, Nn);
  k_aggregate<<<cdiv((long long)Nn * 32, 256), 256, 0, stream>>>(csr, st, deg, ei, E, ev, hA, b1, hB, Nn, 1);

  // ---------------- layer 2 ----------------
  k_gemm_wmma<<<cdiv(Nn, 16), 128, 0, stream>>>(hB, W2, hA, Nn, HID);
  k_scores<<<cdiv((long long)Nn * 32, 256), 256, 0, stream>>>(hA, as2, ad2, scS, scD, Nn);
  k_attn<<<cdiv((long long)Nn * 32, 256), 256, 0, stream>>>(csr, st, deg, ei, E, scS, scD, ev, Nn);
  k_aggregate<<<cdiv((long long)Nn * 32, 256), 256, 0, stream>>>(csr, st, deg, ei, E, ev, hA, b2, hB, Nn, 0);

  // ---------------- pool + head ----------------
  k_init_pool<<<cdiv((long long)G * HID, 256), 256, 0, stream>>>(gsum, gcnt, G);
  k_pool<<<cdiv((long long)Nn * HID, 256), 256, 0, stream>>>(hB, batch, gsum, gcnt, Nn);
  k_head<<<G, 32, 0, stream>>>(gsum, gcnt, lw, lb, cw, cb, out);
}